// LCC_5299989643317
// MI455X (gfx1250) — compile-verified
//
#include <hip/hip_runtime.h>
#include <hip/hip_bf16.h>

// LCC (9x9 local cross-correlation) loss for 16x1x1024x1024 fp32 pairs.
//
// Memory-bound: 128 MB input @ 23.3 TB/s => ~5.5 us floor. Whole pipeline
// (5 box sums -> cc -> mean) folded into one pass: 64x64 output tiles with a
// 72x72 halo in LDS (1.27x read amplification), scalar out, no HBM temps.
// Tile staging uses aligned 128-bit global loads + 128-bit LDS stores
// (c0-4 is 0 mod 4, so every halo row is exactly 18 aligned float4s, each
// fully in- or out-of-bounds).
//
// Horizontal 9-wide box sum is an exact fp32 GEMM against a constant 0/1 band
// matrix using V_WMMA_F32_16X16X4_F32 (weights are exactly 1.0, fp32
// accumulation => exact per-op adds; bf16/f16 WMMA would wreck the
// cancellation-prone variance terms). Vertical 9-tall box sum is an O(1)/row
// running sum on VALU. Band-matrix B operands are hoisted out of the field
// loop (12 VGPRs) so the hot loop is WMMA + ds_load only.

typedef float v2f __attribute__((ext_vector_type(2)));
typedef float v8f __attribute__((ext_vector_type(8)));

#define IMG 1024
#define TILE 64        // output tile edge
#define HREG 72        // halo region edge = TILE + 8
#define LSTR 76        // LDS row stride: mult of 4 (b128 stores); 12m mod 64 -> 16 distinct banks
#define ROWV4 18       // 72 floats per halo row = 18 aligned float4s
#define WIN_INV (1.0f / 81.0f)
#define EPSV 1e-5f
#define NPIX (16.0f * 1024.0f * 1024.0f)

__global__ __launch_bounds__(256) void lcc_tile_kernel(
    const float* __restrict__ I, const float* __restrict__ J,
    float* __restrict__ partials)
{
    __shared__ __align__(16) float sI[HREG * LSTR];   // 72x72 I halo tile
    __shared__ __align__(16) float sJ[HREG * LSTR];   // 72x72 J halo tile
    __shared__ __align__(16) float sV[TILE * LSTR];   // 64x72 vertical box sums (reused per field)
    __shared__ float sWave[8];

    const int tid  = threadIdx.x;
    const int lane = tid & 31;
    const int wid  = tid >> 5;

    const int r0 = blockIdx.y * TILE;
    const int c0 = blockIdx.x * TILE;
    const size_t base = (size_t)blockIdx.z * (IMG * IMG);

    // ---- Stage halo tiles with 128-bit loads/stores.
    // Each float4 is entirely in-bounds or entirely out (zero-fill = padding).
    for (int i = tid; i < HREG * ROWV4; i += 256) {
        int u = i / ROWV4, j = i - u * ROWV4;
        int gr = r0 + u - 4;
        int gc = c0 - 4 + 4 * j;                  // aligned: (c0-4) % 4 == 0
        float4 iv = {0.0f, 0.0f, 0.0f, 0.0f};
        float4 jv = {0.0f, 0.0f, 0.0f, 0.0f};
        if (gr >= 0 && gr < IMG && gc >= 0 && gc <= IMG - 4) {
            size_t g = base + (size_t)gr * IMG + gc;
            iv = *(const float4*)(I + g);
            jv = *(const float4*)(J + g);
        }
        *(float4*)(sI + u * LSTR + 4 * j) = iv;
        *(float4*)(sJ + u * LSTR + 4 * j) = jv;
    }
    __syncthreads();

    // per-field raw value (products formed on the fly; q resolved at compile time)
    auto rawq = [&](int q, int u, int v) -> float {
        float a = sI[u * LSTR + v];
        float b = sJ[u * LSTR + v];
        switch (q) {
            case 0:  return a;
            case 1:  return b;
            case 2:  return a * a;
            case 3:  return b * b;
            default: return a * b;
        }
    };

    // WMMA lane decomposition (32-bit 16x4 A / 4x16 B / 16x16 C-D layouts):
    //   lanes 0-15 hold K = k0, k0+1 ; lanes 16-31 hold K = k0+2, k0+3
    const int m     = lane & 15;   // A row (M) / B-D column (N)
    const int khalf = lane >> 4;

    // Hoisted band-matrix B operands: T[k][j] = (j <= k <= j+8), j = m.
    v2f bmat[6];
#pragma unroll
    for (int kc = 0; kc < 6; ++kc) {
        const int k0 = kc * 4 + khalf * 2;
        bmat[kc].x = (k0     >= m && k0     <= m + 8) ? 1.0f : 0.0f;
        bmat[kc].y = (k0 + 1 >= m && k0 + 1 <= m + 8) ? 1.0f : 0.0f;
    }

    // Accumulators: [subtile-of-wave][field], each 16x16 f32 = 8 VGPRs.
    v8f acc[2][5];
#pragma unroll
    for (int s = 0; s < 2; ++s)
#pragma unroll
        for (int q = 0; q < 5; ++q) {
            v8f z = {};
            acc[s][q] = z;
        }

#pragma unroll
    for (int q = 0; q < 5; ++q) {
        // ---- Vertical 9-tall running box sum: V[r][v] = sum_{u=r..r+8} raw[u][v]
        // 2 threads per column (rows 0..31 / 32..63), O(1) per row after init.
        if (tid < 2 * HREG) {
            int v  = tid >> 1;
            int rs = (tid & 1) * 32;
            float s = 0.0f;
#pragma unroll
            for (int u = 0; u < 9; ++u) s += rawq(q, rs + u, v);
            for (int r = rs; r < rs + 32; ++r) {
                sV[r * LSTR + v] = s;
                if (r + 9 < HREG) s += rawq(q, r + 9, v) - rawq(q, r, v);
            }
        }
        __syncthreads();

        // ---- Horizontal 9-wide box sum as band-matrix GEMM on the matrix pipe:
        // Out(16x16) = V(16x24) x T(24x16), 6 accumulating WMMAs per subtile.
#pragma unroll
        for (int i = 0; i < 2; ++i) {
            const int st = wid * 2 + i;          // subtile 0..15 in the 4x4 grid
            const int sr = st >> 2, sc = st & 3;
            const float* vrow = &sV[(sr * 16 + m) * LSTR + sc * 16];
#pragma unroll
            for (int kc = 0; kc < 6; ++kc) {
                const int k0 = kc * 4 + khalf * 2;
                v2f a;
                a.x = vrow[k0];
                a.y = vrow[k0 + 1];
                acc[i][q] = __builtin_amdgcn_wmma_f32_16x16x4_f32(
                    false, a, false, bmat[kc], (short)0, acc[i][q], false, false);
            }
        }
        __syncthreads();   // protect sV before next field overwrites it
    }

    // ---- Pointwise cc + local accumulation (each lane owns 8 pixels/subtile)
    float lsum = 0.0f;
#pragma unroll
    for (int i = 0; i < 2; ++i) {
#pragma unroll
        for (int p = 0; p < 8; ++p) {
            float Is  = acc[i][0][p];
            float Js  = acc[i][1][p];
            float I2s = acc[i][2][p];
            float J2s = acc[i][3][p];
            float IJs = acc[i][4][p];
            float cross = IJs - Is * Js * WIN_INV;
            float Iv    = I2s - Is * Is * WIN_INV;
            float Jv    = J2s - Js * Js * WIN_INV;
            float cc    = cross * cross / (Iv * Jv + EPSV);
            lsum += cc;
        }
    }

    // ---- Deterministic reduction: wave32 butterfly -> LDS -> one partial/WG
    for (int off = 16; off > 0; off >>= 1)
        lsum += __shfl_xor(lsum, off, 32);
    if (lane == 0) sWave[wid] = lsum;
    __syncthreads();
    if (tid == 0) {
        float t = 0.0f;
#pragma unroll
        for (int w = 0; w < 8; ++w) t += sWave[w];
        partials[(blockIdx.z * 16 + blockIdx.y) * 16 + blockIdx.x] = t;
    }
}

__global__ __launch_bounds__(256) void lcc_reduce_kernel(
    const float* __restrict__ partials, float* __restrict__ out, int n)
{
    __shared__ float s[256];
    float t = 0.0f;
    for (int i = threadIdx.x; i < n; i += 256) t += partials[i];
    s[threadIdx.x] = t;
    __syncthreads();
    for (int off = 128; off > 0; off >>= 1) {
        if (threadIdx.x < off) s[threadIdx.x] += s[threadIdx.x + off];
        __syncthreads();
    }
    if (threadIdx.x == 0) out[0] = 1.0f - s[0] * (1.0f / NPIX);
}

extern "C" void kernel_launch(void* const* d_in, const int* in_sizes, int n_in,
                              void* d_out, int out_size, void* d_ws, size_t ws_size,
                              hipStream_t stream) {
    const float* I = (const float*)d_in[0];
    const float* J = (const float*)d_in[1];
    float* out      = (float*)d_out;
    float* partials = (float*)d_ws;   // 4096 floats = 16 KB scratch

    dim3 grid(IMG / TILE, IMG / TILE, 16);   // 16 x 16 x 16 = 4096 workgroups
    lcc_tile_kernel<<<grid, 256, 0, stream>>>(I, J, partials);
    lcc_reduce_kernel<<<1, 256, 0, stream>>>(partials, out, 16 * 16 * 16);
}